// LogNeuralCDE_79173427134599
// MI455X (gfx1250) — compile-verified
//
#include <hip/hip_runtime.h>
#include <math.h>

// CDNA5 (gfx1250) wave32 WMMA implementation of a log-ODE Neural CDE.
// B=256 batch split into 16 workgroups of 16; all matmuls are 16x16 tiles fed
// by V_WMMA_F32_16X16X4_F32 (full f32 precision through the 32-step Heun chain).

typedef float v2f __attribute__((ext_vector_type(2)));
typedef float v8f __attribute__((ext_vector_type(8)));

#if defined(__HIP_DEVICE_COMPILE__)
#if !__has_builtin(__builtin_amdgcn_wmma_f32_16x16x4_f32)
#error "gfx1250 f32 WMMA builtin not available"
#endif
#endif

#define SK 68   // LDS dword stride for weight rows: A-frag b64 loads are bank-conflict-free
#define SN 18   // LDS dword stride for activation rows: B-frag reads / C-tile writes conflict-free

__device__ __forceinline__ v8f wmma4(v2f a, v2f b, v8f c) {
  // D(16x16,f32) += A(16x4,f32) * B(4x16,f32)
  return __builtin_amdgcn_wmma_f32_16x16x4_f32(false, a, false, b, (short)0, c, false, false);
}

// A-fragment: lane (m = lane&15, hi = lane>>4) holds A[m][k0+2*hi + {0,1}]
__device__ __forceinline__ v2f ldA(const float* w, int stride, int row0, int k0, int m, int hi) {
  const float* p = w + (row0 + m) * stride + k0 + 2 * hi;
  v2f r; r.x = p[0]; r.y = p[1]; return r;
}

// B-fragment: lane (n = lane&15, hi) holds B[k0+2*hi + {0,1}][n], from [k][n] LDS layout
__device__ __forceinline__ v2f ldB(const float* act, int r0, int k0, int n, int hi) {
  int k = r0 + k0 + 2 * hi;
  v2f r; r.x = act[k * SN + n]; r.y = act[(k + 1) * SN + n]; return r;
}

// C = W[wrow0:+16, 0:64] @ act[arow0:+64, 0:16]   (16 chained K=4 WMMAs)
__device__ __forceinline__ v8f mm64(const float* w, int wrow0, const float* act, int arow0,
                                    int m, int n, int hi) {
  v8f c = {};
#pragma unroll
  for (int kk = 0; kk < 16; ++kk)
    c = wmma4(ldA(w, SK, wrow0, kk * 4, m, hi), ldB(act, arow0, kk * 4, n, hi), c);
  return c;
}

__global__ __launch_bounds__(256)
void logcde_kernel(const float* __restrict__ ts, const float* __restrict__ intervals,
                   const float* __restrict__ logsig, const float* __restrict__ x0,
                   const float* __restrict__ W1, const float* __restrict__ b1,
                   const float* __restrict__ W2, const float* __restrict__ b2,
                   const float* __restrict__ Wv0, const float* __restrict__ bv0,
                   const float* __restrict__ Wv1, const float* __restrict__ bv1,
                   const float* __restrict__ Wvo, const float* __restrict__ bvo,
                   float* __restrict__ out)
{
  // ---- LDS (about 305 KB; CDNA5 allows up to 320 KB per workgroup) ----
  __shared__ float sWv0[64 * SK], sWv1[64 * SK], sWvo[512 * SK];
  __shared__ float sW1[64 * 10], sW2[16 * SK];
  __shared__ float sBv0[64], sBv1[64], sBvo[512], sB1[64], sB2[16];
  __shared__ float Ybuf[64 * SN], Y2buf[64 * SN];
  __shared__ float h0b[64 * SN], sp0b[64 * SN], h1b[64 * SN], sp1b[64 * SN];
  __shared__ float dh0g0[64 * SN], dh0g1[64 * SN], dh1g0[64 * SN], dh1g1[64 * SN];
  __shared__ float drv1[64 * SN];
  __shared__ float vfo[512 * SN], tpb[512 * SN];
  __shared__ float lsb[16 * 38];
  __shared__ float x0b[8 * SN];
  __shared__ int   segtab[33];
  __shared__ float rsctab[33];
  __shared__ float logitsb[10 * SN];
  __shared__ float sDt[1];

  const int tid  = threadIdx.x;
  const int lane = tid & 31;
  const int n    = lane & 15;   // batch column within tile (and A-row m)
  const int hi   = lane >> 4;
  const int wave = tid >> 5;    // 0..7
  const int g    = wave >> 2;   // direction-parity group (0/1)
  const int w4   = wave & 3;    // 16-row output tile owner
  const int m    = n;
  const int wgBase = blockIdx.x * 16;

  // ---- stage weights / biases / x0 tile into LDS ----
  for (int e = tid; e < 64 * 64; e += 256) {
    sWv0[(e >> 6) * SK + (e & 63)] = Wv0[e];
    sWv1[(e >> 6) * SK + (e & 63)] = Wv1[e];
  }
  for (int e = tid; e < 512 * 64; e += 256) sWvo[(e >> 6) * SK + (e & 63)] = Wvo[e];
  for (int e = tid; e < 64 * 8; e += 256) sW1[(e >> 3) * 10 + (e & 7)] = W1[e];
  for (int e = tid; e < 16 * SK; e += 256) sW2[e] = 0.f;  // zero-pad rows 10..15
  for (int e = tid; e < 64; e += 256) { sBv0[e] = bv0[e]; sBv1[e] = bv1[e]; sB1[e] = b1[e]; }
  for (int e = tid; e < 512; e += 256) sBvo[e] = bvo[e];
  if (tid < 16) sB2[tid] = (tid < 10) ? b2[tid] : 0.f;
  for (int e = tid; e < 16 * 8; e += 256)
    x0b[(e & 7) * SN + (e >> 3)] = x0[(size_t)(wgBase + (e >> 3)) * 8 + (e & 7)];
  __syncthreads();
  for (int e = tid; e < 10 * 64; e += 256) sW2[(e >> 6) * SK + (e & 63)] = W2[e];
  if (tid == 0) {
    float t0 = ts[0], t1 = ts[32];
    float dtv = (t1 - t0) * (1.f / 32.f);
    sDt[0] = dtv;
    for (int s = 0; s <= 32; ++s) {      // eval times t0 + dt*s, s=0..32
      float t = t0 + dtv * (float)s;
      int i = 0;                          // searchsorted left: first intervals[i] >= t
      while (i < 33 && intervals[i] < t) ++i;
      if (i < 1) i = 1;
      if (i > 32) i = 32;
      segtab[s] = i - 1;
      rsctab[s] = 1.f / (intervals[i] - intervals[i - 1]);
    }
  }
  __syncthreads();
  const float dtv = sDt[0];

  // ---- one evaluation of func(t, y): returns drive/dt tile in C-registers (group 0) ----
  auto func_eval = [&](const float* inY, int entry) -> v8f {
    const int   seg = segtab[entry];
    const float rsc = rsctab[entry];
    // stage this segment's logsig coefficients for the 16 batch columns
    for (int e = tid; e < 16 * 37; e += 256) {
      int nn = e / 37, j = e - nn * 37;
      lsb[nn * 38 + j] = logsig[((size_t)(wgBase + nn) * 32 + (size_t)seg) * 37 + j];
    }
    __syncthreads();
    // stage 1: u0 = Wv0@y + bv0 ; h0 = silu(u0), sp0 = silu'(u0)
    if (g == 0) {
      v8f c = mm64(sWv0, w4 * 16, inY, 0, m, n, hi);
#pragma unroll
      for (int i = 0; i < 8; ++i) {
        int r = w4 * 16 + i + 8 * hi;
        float u = c[i] + sBv0[r];
        float sg = 1.f / (1.f + expf(-u));
        h0b[r * SN + n]  = u * sg;
        sp0b[r * SN + n] = sg * (1.f + u * (1.f - sg));
      }
    }
    __syncthreads();
    // stage 2: h1 = silu(Wv1@h0 + bv1), sp1
    if (g == 0) {
      v8f c = mm64(sWv1, w4 * 16, h0b, 0, m, n, hi);
#pragma unroll
      for (int i = 0; i < 8; ++i) {
        int r = w4 * 16 + i + 8 * hi;
        float u = c[i] + sBv1[r];
        float sg = 1.f / (1.f + expf(-u));
        h1b[r * SN + n]  = u * sg;
        sp1b[r * SN + n] = sg * (1.f + u * (1.f - sg));
      }
    }
    __syncthreads();
    // stage 3: vf_out = tanh(Wvo@h1 + bvo) [512x16], tanh' ; drive += ls[1+b]*vf_out
    v8f drive = {};
#pragma unroll 1
    for (int bb = 0; bb < 4; ++bb) {
      int b = bb * 2 + g;
      v8f c = mm64(sWvo, b * 64 + w4 * 16, h1b, 0, m, n, hi);
      float cb = lsb[n * 38 + 1 + b];
#pragma unroll
      for (int i = 0; i < 8; ++i) {
        int r = b * 64 + w4 * 16 + i + 8 * hi;
        float th = tanhf(c[i] + sBvo[r]);
        vfo[r * SN + n] = th;
        tpb[r * SN + n] = 1.f - th * th;
        drive[i] += cb * th;
      }
    }
    __syncthreads();
    // stage 4: JVPs along the 8 columns of vf_out; two directions in flight (one per group)
    float* dh0 = g ? dh0g1 : dh0g0;
    float* dh1 = g ? dh1g1 : dh1g0;
#pragma unroll 1
    for (int dd = 0; dd < 4; ++dd) {
      int d = dd * 2 + g;
      {
        v8f c = mm64(sWv0, w4 * 16, vfo, d * 64, m, n, hi);  // tangent = vf_out rows 64d..
#pragma unroll
        for (int i = 0; i < 8; ++i) {
          int r = w4 * 16 + i + 8 * hi;
          dh0[r * SN + n] = sp0b[r * SN + n] * c[i];
        }
      }
      __syncthreads();
      {
        v8f c = mm64(sWv1, w4 * 16, dh0, 0, m, n, hi);
#pragma unroll
        for (int i = 0; i < 8; ++i) {
          int r = w4 * 16 + i + 8 * hi;
          dh1[r * SN + n] = sp1b[r * SN + n] * c[i];
        }
      }
      __syncthreads();
#pragma unroll 1
      for (int b = 0; b < 8; ++b) {   // jvps[d][b]; antisymmetric Hall-pair contraction
        if (b == d) continue;         // wave-uniform skip
        v8f c = mm64(sWvo, b * 64 + w4 * 16, dh1, 0, m, n, hi);
        float s;
        if (d < b) s =  lsb[n * 38 + 9 + (7 * d - (d * (d - 1)) / 2) + (b - d - 1)];
        else       s = -lsb[n * 38 + 9 + (7 * b - (b * (b - 1)) / 2) + (d - b - 1)];
#pragma unroll
        for (int i = 0; i < 8; ++i) {
          int r = b * 64 + w4 * 16 + i + 8 * hi;
          drive[i] += s * tpb[r * SN + n] * c[i];
        }
      }
      __syncthreads();
    }
    // combine the two groups' drive partials into group 0's registers
    if (g == 1) {
#pragma unroll
      for (int i = 0; i < 8; ++i) drv1[(w4 * 16 + i + 8 * hi) * SN + n] = drive[i];
    }
    __syncthreads();
    if (g == 0) {
#pragma unroll
      for (int i = 0; i < 8; ++i)
        drive[i] = (drive[i] + drv1[(w4 * 16 + i + 8 * hi) * SN + n]) * rsc;
    }
    return drive;
  };

  // ---- y0 = W1 @ x0 + b1 (K=8 -> 2 WMMAs per tile) ----
  v8f y = {};
  if (g == 0) {
    v8f c = {};
#pragma unroll
    for (int kk = 0; kk < 2; ++kk)
      c = wmma4(ldA(sW1, 10, w4 * 16, kk * 4, m, hi), ldB(x0b, 0, kk * 4, n, hi), c);
#pragma unroll
    for (int i = 0; i < 8; ++i) {
      int r = w4 * 16 + i + 8 * hi;
      y[i] = c[i] + sB1[r];
      Ybuf[r * SN + n] = y[i];
    }
  }
  __syncthreads();

  // ---- 32 fixed Heun steps ----
#pragma unroll 1
  for (int s = 0; s < 32; ++s) {
    v8f k1 = func_eval(Ybuf, s);
    if (g == 0) {
#pragma unroll
      for (int i = 0; i < 8; ++i)
        Y2buf[(w4 * 16 + i + 8 * hi) * SN + n] = y[i] + dtv * k1[i];
    }
    __syncthreads();
    v8f k2 = func_eval(Y2buf, s + 1);
    if (g == 0) {
#pragma unroll
      for (int i = 0; i < 8; ++i) {
        y[i] += 0.5f * dtv * (k1[i] + k2[i]);
        Ybuf[(w4 * 16 + i + 8 * hi) * SN + n] = y[i];
      }
    }
    __syncthreads();
  }

  // ---- logits = W2 @ yT + b2 (one 16x16 tile, rows 10..15 zero-padded), softmax ----
  if (wave == 0) {
    v8f c = mm64(sW2, 0, Ybuf, 0, m, n, hi);
#pragma unroll
    for (int i = 0; i < 8; ++i) {
      int r = i + 8 * hi;
      if (r < 10) logitsb[r * SN + n] = c[i] + sB2[r];
    }
  }
  __syncthreads();
  if (tid < 16) {
    float v[10];
    float mx = -1e30f;
#pragma unroll
    for (int j = 0; j < 10; ++j) { v[j] = logitsb[j * SN + tid]; mx = fmaxf(mx, v[j]); }
    float sum = 0.f;
#pragma unroll
    for (int j = 0; j < 10; ++j) { v[j] = expf(v[j] - mx); sum += v[j]; }
    float inv = 1.f / sum;
#pragma unroll
    for (int j = 0; j < 10; ++j) out[(size_t)(wgBase + tid) * 10 + j] = v[j] * inv;
  }
}

extern "C" void kernel_launch(void* const* d_in, const int* in_sizes, int n_in,
                              void* d_out, int out_size, void* d_ws, size_t ws_size,
                              hipStream_t stream) {
  const float* ts        = (const float*)d_in[0];
  const float* intervals = (const float*)d_in[1];
  const float* logsig    = (const float*)d_in[2];
  const float* x0        = (const float*)d_in[3];
  // d_in[4] = pairs (int32): Hall-pair order (i<j row-major) is hardcoded to match the reference
  const float* W1  = (const float*)d_in[5];
  const float* b1  = (const float*)d_in[6];
  const float* W2  = (const float*)d_in[7];
  const float* b2  = (const float*)d_in[8];
  const float* Wv0 = (const float*)d_in[9];
  const float* bv0 = (const float*)d_in[10];
  const float* Wv1 = (const float*)d_in[11];
  const float* bv1 = (const float*)d_in[12];
  const float* Wvo = (const float*)d_in[13];
  const float* bvo = (const float*)d_in[14];
  (void)in_sizes; (void)n_in; (void)out_size; (void)d_ws; (void)ws_size;
  logcde_kernel<<<dim3(16), dim3(256), 0, stream>>>(
      ts, intervals, logsig, x0, W1, b1, W2, b2, Wv0, bv0, Wv1, bv1, Wvo, bvo, (float*)d_out);
}